// NNFBP_19439021982126
// MI455X (gfx1250) — compile-verified
//
#include <hip/hip_runtime.h>
#include <math.h>

// ---------------- geometry constants (match reference) ----------------
#define NANG 720
#define NDET 1024
#define HH   512
#define WW   512
#define NB   8
#define KCONV 11
#define NCH  (NANG / 16)   // 45 chunks of 16 angles

// DS = 4/1024, DX = 2/512  ->  DX/DS == 1.0 exactly.
__device__ __constant__ float kDS  = 4.0f / 1024.0f;
__device__ __constant__ float kS0  = -2.0f + (4.0f / 1024.0f) * 0.5f;
__device__ __constant__ float kDX  = 2.0f / 512.0f;
__device__ __constant__ float kDTH = 3.14159265358979323846f / 720.0f;

typedef float v2f __attribute__((ext_vector_type(2)));
typedef float v8f __attribute__((ext_vector_type(8)));

#if __has_builtin(__builtin_amdgcn_global_load_async_to_lds_b32)
#define HAS_ASYNC_LDS 1
#endif

#ifdef HAS_ASYNC_LDS
__device__ __forceinline__ void async_ld_f32(const float* gsrc, float* ldst) {
  __builtin_amdgcn_global_load_async_to_lds_b32(
      (__attribute__((address_space(1))) int*)(void*)gsrc,
      (__attribute__((address_space(3))) int*)(void*)ldst, 0, 0);
}
__device__ __forceinline__ void wait_async3() {
#if __has_builtin(__builtin_amdgcn_s_wait_asynccnt)
  __builtin_amdgcn_s_wait_asynccnt(3);
#else
  asm volatile("s_wait_asynccnt 0x3" ::: "memory");
#endif
}
__device__ __forceinline__ void wait_async0() {
#if __has_builtin(__builtin_amdgcn_s_wait_asynccnt)
  __builtin_amdgcn_s_wait_asynccnt(0);
#else
  asm volatile("s_wait_asynccnt 0x0" ::: "memory");
#endif
}
#endif

// =====================================================================
// Kernel 1: detector-axis conv (left pad K-1) as a banded matmul on
// V_WMMA_F32_16X16X4_F32 (16 rows x 16 dets per wave, 7 K-steps of 4).
// =====================================================================
__global__ __launch_bounds__(256)
void fbp_conv_wmma(const float* __restrict__ x,
                   const float* __restrict__ w,
                   float* __restrict__ y) {
  const int tx   = threadIdx.x;
  const int lane = tx & 31;
  const int wid  = tx >> 5;
  const int rowBase = blockIdx.y * 16;     // 360 row-groups of 16 (B*A = 5760)
  const int D0      = blockIdx.x * 128;    // 8 det-groups of 128

  __shared__ float xs[16 * 140];           // stride 140 -> 32 distinct banks on col reads

  for (int t = tx; t < 16 * 140; t += 256) {
    int m = t / 140, col = t % 140;
    int d = D0 - 10 + col;
    float v = 0.0f;
    if (d >= 0 && d < NDET) v = x[(size_t)(rowBase + m) * NDET + d];
    xs[t] = v;
  }
  __syncthreads();

  // ISA 7.12.2 wave32 layouts:
  //  A 16x4 f32: VGPR0 lanes0-15 K=0 / lanes16-31 K=2 ; VGPR1: K=1 / K=3
  //  C 16x16 f32: VGPR r: lanes0-15 M=r, lanes16-31 M=8+r; N = lane%16
  const int n  = lane & 15;
  const int kh = (lane >> 4) << 1;
  const int m  = lane & 15;

  v2f bfrag[7];
#pragma unroll
  for (int c = 0; c < 7; ++c) {
#pragma unroll
    for (int v = 0; v < 2; ++v) {
      int K  = 4 * c + kh + v;
      int kk = K - n;
      bfrag[c][v] = (kk >= 0 && kk < KCONV) ? w[kk] : 0.0f;
    }
  }

  const int base = 16 * wid;
  v8f acc = {0.f, 0.f, 0.f, 0.f, 0.f, 0.f, 0.f, 0.f};

#if __has_builtin(__builtin_amdgcn_wmma_f32_16x16x4_f32)
#pragma unroll
  for (int c = 0; c < 7; ++c) {
    v2f afrag;
    afrag.x = xs[m * 140 + base + 4 * c + kh + 0];
    afrag.y = xs[m * 140 + base + 4 * c + kh + 1];
    acc = __builtin_amdgcn_wmma_f32_16x16x4_f32(false, afrag, false, bfrag[c],
                                                (short)0, acc, false, false);
  }
#else
  {
    const int mb = (lane >> 4) * 8;
#pragma unroll
    for (int r = 0; r < 8; ++r) {
      float s = 0.0f;
      int mm = mb + r;
      for (int k = 0; k < KCONV; ++k)
        s = fmaf(w[k], xs[mm * 140 + base + n + k], s);
      acc[r] = s;
    }
  }
#endif

  const int mb = (lane >> 4) * 8;
#pragma unroll
  for (int r = 0; r < 8; ++r) {
    y[(size_t)(rowBase + mb + r) * NDET + (D0 + 16 * wid + n)] = acc[r];
  }
}

// =====================================================================
// Kernel 2: backprojection with double-buffered ASYNC global->LDS staging.
// Block = 256 thr, tile = 32(i) x 32(j) pixels, one batch per blockIdx.z.
// Thread: j = j0 + (tx&31); i = i0 + (tx>>5) + 8*q, q = 0..3.
// Per 16-angle chunk: 16x48-float segments; DX/DS == 1 so u advances by
// exactly cos/sin per pixel step; u provably in range (|t| <= 1.42 < 2).
// =====================================================================
__global__ __launch_bounds__(256)
void fbp_backproj(const float* __restrict__ y, float* __restrict__ out) {
  const int tx  = threadIdx.x;
  const int b   = blockIdx.z;
  const int i0  = blockIdx.y * 32;
  const int j0  = blockIdx.x * 32;
  const int jj  = tx & 31;
  const int wid = tx >> 5;

  __shared__ float lc[2][16], ls[2][16], lu[2][16];
  __shared__ int   ldb[2][16];
  __shared__ float seg[2][16 * 48];       // 6 KB double buffer (span <= 46)

  float acc[4] = {0.f, 0.f, 0.f, 0.f};

  const float pxi0 = -1.0f + ((float)i0 + 0.5f) * kDX;
  const float pxj0 = -1.0f + ((float)j0 + 0.5f) * kDX;
  const float invDS = 1.0f / kDS;
  const float* __restrict__ ybat = y + (size_t)b * NANG * NDET;

  // ---- per-chunk angle constants: threads 0..15, chunk ch -> buffer bf
  auto make_consts = [&](int ch, int bf) {
    if (tx < 16) {
      float th = ((float)(ch * 16 + tx) + 0.5f) * kDTH;
      float st, ct;
      sincosf(th, &st, &ct);
      float u00  = (pxi0 * ct + pxj0 * st - kS0) * invDS;
      float umin = u00 + fminf(0.0f, 31.0f * ct) + fminf(0.0f, 31.0f * st);
      int   db   = (int)floorf(umin) - 1;        // -1: fp rounding margin
      lc[bf][tx]  = ct;
      ls[bf][tx]  = st;
      lu[bf][tx]  = u00 - (float)db;
      ldb[bf][tx] = db;
    }
  };

#ifdef HAS_ASYNC_LDS
  // ---- issue 3 async b32 loads per thread into buffer bf for chunk ch
  auto issue_seg = [&](int ch, int bf) {
    const float* __restrict__ yrow = ybat + (size_t)(ch * 16) * NDET;
#pragma unroll
    for (int r = 0; r < 3; ++r) {
      int t  = tx + 256 * r;
      int al = t / 48, s2 = t % 48;
      int d  = ldb[bf][al] + s2;
      d = min(max(d, 0), NDET - 1);             // provably in range; fp insurance
      async_ld_f32(yrow + (size_t)al * NDET + d, &seg[bf][al * 48 + s2]);
    }
  };

  make_consts(0, 0);
  __syncthreads();
  issue_seg(0, 0);

  for (int c = 0; c < NCH; ++c) {
    const int cb = c & 1, nb = cb ^ 1;
    __syncthreads();                            // chunk c-1 compute done (seg[nb] free)
    if (c + 1 < NCH) make_consts(c + 1, nb);
    __syncthreads();                            // consts[nb] visible to all
    if (c + 1 < NCH) { issue_seg(c + 1, nb); wait_async3(); }
    else             { wait_async0(); }
    __syncthreads();                            // all waves' seg[cb] deposits landed
#else
  for (int c = 0; c < NCH; ++c) {
    const int cb = 0;
    __syncthreads();
    make_consts(c, cb);
    __syncthreads();
    {
      const float* __restrict__ yrow = ybat + (size_t)(c * 16) * NDET;
      for (int t = tx; t < 16 * 48; t += 256) {
        int al = t / 48, s2 = t % 48;
        int d  = ldb[cb][al] + s2;
        float v = 0.0f;
        if (d >= 0 && d < NDET) v = yrow[(size_t)al * NDET + d];
        seg[cb][al * 48 + s2] = v;
      }
    }
    __syncthreads();
#endif

#pragma unroll 4
    for (int al = 0; al < 16; ++al) {
      const float ct = lc[cb][al], st = ls[cb][al], uo = lu[cb][al];
      const float* __restrict__ sp = &seg[cb][al * 48];
      float u = uo + (float)jj * st + (float)wid * ct;   // i stride 8 between q's
#pragma unroll
      for (int q = 0; q < 4; ++q) {
        float fi = floorf(u);
        int  idx = (int)fi;                    // in [0, 46] by construction
        float f  = u - fi;
        float v0 = sp[idx];
        float v1 = sp[idx + 1];
        acc[q] = fmaf(f, v1 - v0, acc[q] + v0);
        u += 8.0f * ct;
      }
    }
  }

  const float scale = kDTH;
  const size_t o = ((size_t)b * HH + (i0 + wid)) * WW + (j0 + jj);
#pragma unroll
  for (int q = 0; q < 4; ++q)
    out[o + (size_t)(8 * q) * WW] = acc[q] * scale;
}

// =====================================================================
extern "C" void kernel_launch(void* const* d_in, const int* in_sizes, int n_in,
                              void* d_out, int out_size, void* d_ws, size_t ws_size,
                              hipStream_t stream) {
  const float* x = (const float*)d_in[0];   // [8,1,720,1024] f32
  const float* w = (const float*)d_in[1];   // [1,1,1,11]     f32
  float* out = (float*)d_out;               // [8,1,512,512]  f32
  float* y   = (float*)d_ws;                // filtered sino: 8*720*1024 f32 (23.6 MB)

  fbp_conv_wmma<<<dim3(8, 360, 1), 256, 0, stream>>>(x, w, y);
  fbp_backproj<<<dim3(16, 16, 8), 256, 0, stream>>>(y, out);
}